// GraphEncoder_4037269258803
// MI455X (gfx1250) — compile-verified
//
#include <hip/hip_runtime.h>
#include <hip/hip_bf16.h>
#include <math.h>

// ---------------------------------------------------------------------------
// GraphEncoder forward for MI455X (gfx1250, wave32, WMMA)
// Dense GEMMs -> V_WMMA_F32_16X16X4_F32, B staged via LDS (pair-packed so a
// lane's two K-values are one ds_load_b64). Workload is bound by edge-scatter
// HBM traffic, so exact-f32 matrix math is free.
// ---------------------------------------------------------------------------

#define N_NODES 50000
#define N_EDGES 500000
#define N_FEAT  128
#define HID     256
#define LAT     64
#define N_GR    512
#define HEADS   8
#define DH      32
#define EPSV    1e-5f
#define KC      64          // K-chunk staged in LDS

typedef __attribute__((ext_vector_type(2))) float v2f;
typedef __attribute__((ext_vector_type(8))) float v8f;

// ---------------------------------------------------------------------------
// f32 WMMA GEMM: C[M,N] = A[M,K] @ B[K,N] (+bias)(+=C)(relu)
// Block: 128 threads = 4 waves; block tile 64(M) x 64(N); wave tile 16 x 64.
// B tile staged in LDS in 64-row K-chunks, pair-packed:
//   lds[(k>>1)*128 + col*2 + (k&1)] = B[kc+k][n0+col]
// so the WMMA B operand (rows k0+2h, k0+2h+1 at col) is one ds_load_b64.
// A operand: one global_load_b64 per K-step (row m0+r, elems k0+2h, k0+2h+1).
// C/D: VGPR v -> row m0 + v + 8*half, col n0 + 16*t + (lane&15).
// ---------------------------------------------------------------------------
__global__ __launch_bounds__(128) void gemm_wmma_f32(
    const float* __restrict__ A, const float* __restrict__ B,
    float* __restrict__ C, const float* __restrict__ bias,
    int M, int Nn, int K, int accumulate, int relu)
{
    __shared__ float ldsb[(KC / 2) * 128];   // 16 KB pair-packed B chunk

    const int tid  = threadIdx.x;
    const int wave = tid >> 5;
    const int lane = tid & 31;
    const int half = lane >> 4;
    const int r    = lane & 15;
    const int n0   = blockIdx.y * 64;
    const int m0   = blockIdx.x * 64 + wave * 16;
    const bool rowvalid = (m0 < M);          // ragged M: clamp loads, skip store

    const float* __restrict__ arow = A + (size_t)(rowvalid ? (m0 + r) : r) * K;

    v8f acc[4] = {};
    for (int kc = 0; kc < K; kc += KC) {
        __syncthreads();
        // cooperative, coalesced load of B[kc..kc+KC)[n0..n0+64) -> pair-packed LDS
#pragma unroll
        for (int i = 0; i < (KC * 64) / 128; ++i) {   // 32 iters, 128 threads
            int flat = tid + i * 128;
            int kk = flat >> 6;
            int c  = flat & 63;
            ldsb[(kk >> 1) * 128 + c * 2 + (kk & 1)] =
                B[(size_t)(kc + kk) * Nn + n0 + c];
        }
        __syncthreads();
        __builtin_prefetch(arow + kc + KC, 0, 0);     // next A chunk -> global_prefetch

#pragma unroll
        for (int k0 = 0; k0 < KC; k0 += 4) {
            v2f a = *(const v2f*)(arow + kc + k0 + 2 * half);
            const float* __restrict__ bp = ldsb + ((k0 >> 1) + half) * 128;
#pragma unroll
            for (int t = 0; t < 4; ++t) {
                v2f b = *(const v2f*)(bp + (16 * t + r) * 2);
                acc[t] = __builtin_amdgcn_wmma_f32_16x16x4_f32(
                    false, a, false, b, (short)0, acc[t], false, false);
            }
        }
    }

    if (!rowvalid) return;
#pragma unroll
    for (int t = 0; t < 4; ++t) {
        const int col = n0 + 16 * t + r;
        const float bv = bias ? bias[col] : 0.0f;
#pragma unroll
        for (int v = 0; v < 8; ++v) {
            const int row = m0 + v + 8 * half;
            float val = acc[t][v] + bv;
            float* p = C + (size_t)row * Nn + col;
            if (accumulate) val += *p;
            if (relu) val = fmaxf(val, 0.0f);
            *p = val;
        }
    }
}

// ---------------------------------------------------------------------------
// Small utility kernels
// ---------------------------------------------------------------------------
__global__ void fill_kernel(float* __restrict__ p, size_t n, float v) {
    size_t i = (size_t)blockIdx.x * blockDim.x + threadIdx.x;
    size_t stride = (size_t)gridDim.x * blockDim.x;
    for (; i < n; i += stride) p[i] = v;
}

__device__ inline void atomicMaxF(float* addr, float val) {
    unsigned int* ia = (unsigned int*)addr;
    unsigned int old = *ia;
    while (true) {
        float f = __uint_as_float(old);
        if (f >= val) break;
        unsigned int assumed = old;
        old = atomicCAS(ia, assumed, __float_as_uint(val));
        if (old == assumed) break;
    }
}

// degree over edges(+optional self loops): deg[dst] += 1
__global__ void degree_kernel(const int* __restrict__ ei, float* __restrict__ deg,
                              int E, int total) {
    int i = blockIdx.x * blockDim.x + threadIdx.x;
    if (i >= total) return;
    int d = (i < E) ? ei[E + i] : (i - E);
    atomicAdd(&deg[d], 1.0f);
}

__global__ void rsqrt_kernel(const float* __restrict__ deg, float* __restrict__ dis, int n) {
    int i = blockIdx.x * blockDim.x + threadIdx.x;
    if (i < n) dis[i] = rsqrtf(deg[i]);
}

// GCN scatter: out[dst] += xt[src] * dis[src]*dis[dst]  (block per entry, thread per feat)
__global__ __launch_bounds__(HID) void gcn_scatter_kernel(
    const int* __restrict__ ei, const float* __restrict__ xt,
    const float* __restrict__ dis, float* __restrict__ out, int E) {
    int idx = blockIdx.x, f = threadIdx.x;
    int s, d;
    if (idx < E) { s = ei[idx]; d = ei[E + idx]; } else { s = d = idx - E; }
    float coef = dis[s] * dis[d];
    atomicAdd(&out[(size_t)d * HID + f], xt[(size_t)s * HID + f] * coef);
}

// column stats (sum, sumsq) for BatchNorm: coalesced partial accumulation
__global__ __launch_bounds__(HID) void colstat_partial_kernel(
    const float* __restrict__ X, float* __restrict__ sum, float* __restrict__ sumsq, int rows) {
    int f = threadIdx.x;
    float s = 0.0f, ss = 0.0f;
    for (int row = blockIdx.x; row < rows; row += gridDim.x) {
        float v = X[(size_t)row * HID + f];
        s += v; ss += v * v;
    }
    atomicAdd(&sum[f], s);
    atomicAdd(&sumsq[f], ss);
}

__global__ void colstat_finalize_kernel(const float* __restrict__ sum, const float* __restrict__ sumsq,
                                        float* __restrict__ mu, float* __restrict__ rs, int rows) {
    int f = threadIdx.x;
    if (f >= HID) return;
    float m = sum[f] / (float)rows;
    float var = sumsq[f] / (float)rows - m * m;
    mu[f] = m;
    rs[f] = rsqrtf(var + EPSV);
}

// h += relu( (out - mu) * rs * gamma + beta )
__global__ void bn_apply_residual_kernel(float* __restrict__ h, const float* __restrict__ out,
                                         const float* __restrict__ mu, const float* __restrict__ rs,
                                         const float* __restrict__ gamma, const float* __restrict__ beta,
                                         size_t n) {
    size_t i = (size_t)blockIdx.x * blockDim.x + threadIdx.x;
    size_t stride = (size_t)gridDim.x * blockDim.x;
    for (; i < n; i += stride) {
        int f = (int)(i & (HID - 1));
        float y = (out[i] - mu[f]) * rs[f] * gamma[f] + beta[f];
        h[i] += fmaxf(y, 0.0f);
    }
}

// GAT attention coefficients: al[n,head] = sum_d xt[n, head*32+d] * a[head,d]
__global__ void gat_coeff_kernel(const float* __restrict__ xt,
                                 const float* __restrict__ a_src, const float* __restrict__ a_dst,
                                 float* __restrict__ alsrc, float* __restrict__ aldst, int n) {
    int i = blockIdx.x * blockDim.x + threadIdx.x;
    if (i >= n * HEADS) return;
    int node = i >> 3, hd = i & 7;
    const float* xp = xt + (size_t)node * HID + hd * DH;
    const float* as = a_src + hd * DH;
    const float* ad = a_dst + hd * DH;
    float s = 0.0f, d = 0.0f;
#pragma unroll
    for (int j = 0; j < DH; ++j) { s += xp[j] * as[j]; d += xp[j] * ad[j]; }
    alsrc[i] = s;
    aldst[i] = d;
}

__device__ inline float gat_logit(const float* alsrc, const float* aldst, int s, int d, int hd) {
    float x = alsrc[s * HEADS + hd] + aldst[d * HEADS + hd];
    return (x >= 0.0f) ? x : 0.2f * x;   // leaky_relu 0.2
}

__global__ void gat_max_kernel(const int* __restrict__ ei, const float* __restrict__ alsrc,
                               const float* __restrict__ aldst, float* __restrict__ segmax,
                               int E, int total) {
    int i = blockIdx.x * blockDim.x + threadIdx.x;
    if (i >= total * HEADS) return;
    int idx = i >> 3, hd = i & 7;
    int s, d;
    if (idx < E) { s = ei[idx]; d = ei[E + idx]; } else { s = d = idx - E; }
    atomicMaxF(&segmax[d * HEADS + hd], gat_logit(alsrc, aldst, s, d, hd));
}

__global__ void finite_fix_kernel(float* __restrict__ p, size_t n) {
    size_t i = (size_t)blockIdx.x * blockDim.x + threadIdx.x;
    if (i < n && !isfinite(p[i])) p[i] = 0.0f;
}

__global__ void gat_sum_kernel(const int* __restrict__ ei, const float* __restrict__ alsrc,
                               const float* __restrict__ aldst, const float* __restrict__ segmax,
                               float* __restrict__ segsum, int E, int total) {
    int i = blockIdx.x * blockDim.x + threadIdx.x;
    if (i >= total * HEADS) return;
    int idx = i >> 3, hd = i & 7;
    int s, d;
    if (idx < E) { s = ei[idx]; d = ei[E + idx]; } else { s = d = idx - E; }
    float e = expf(gat_logit(alsrc, aldst, s, d, hd) - segmax[d * HEADS + hd]);
    atomicAdd(&segsum[d * HEADS + hd], e);
}

__global__ __launch_bounds__(HID) void gat_scatter_kernel(
    const int* __restrict__ ei, const float* __restrict__ xt,
    const float* __restrict__ alsrc, const float* __restrict__ aldst,
    const float* __restrict__ segmax, const float* __restrict__ segsum,
    float* __restrict__ out, int E) {
    int idx = blockIdx.x, f = threadIdx.x;
    int s, d;
    if (idx < E) { s = ei[idx]; d = ei[E + idx]; } else { s = d = idx - E; }
    int hd = f >> 5;
    float m = segmax[d * HEADS + hd];
    float e = expf(gat_logit(alsrc, aldst, s, d, hd) - m);
    float alpha = e / fmaxf(segsum[d * HEADS + hd], 1e-16f);
    atomicAdd(&out[(size_t)d * HID + f], xt[(size_t)s * HID + f] * alpha);
}

// SAGE neighbor sum (no self-loops): nbsum[dst] += h[src]
__global__ __launch_bounds__(HID) void sage_scatter_kernel(
    const int* __restrict__ ei, const float* __restrict__ h, float* __restrict__ nbsum, int E) {
    int idx = blockIdx.x, f = threadIdx.x;
    int s = ei[idx], d = ei[E + idx];
    atomicAdd(&nbsum[(size_t)d * HID + f], h[(size_t)s * HID + f]);
}

__global__ void sage_mean_kernel(float* __restrict__ nbsum, const float* __restrict__ cnt, size_t n) {
    size_t i = (size_t)blockIdx.x * blockDim.x + threadIdx.x;
    size_t stride = (size_t)gridDim.x * blockDim.x;
    for (; i < n; i += stride) nbsum[i] /= fmaxf(cnt[i / HID], 1.0f);
}

// gate[n] = relu_mlp_hidden[n] . Wg2 + bg2
__global__ void gate_kernel(const float* __restrict__ g1, const float* __restrict__ Wg2,
                            const float* __restrict__ bg2, float* __restrict__ gate, int n) {
    int node = blockIdx.x * blockDim.x + threadIdx.x;
    if (node >= n) return;
    const float* p = g1 + (size_t)node * (HID / 2);
    float s = bg2[0];
    for (int j = 0; j < HID / 2; ++j) s += p[j] * Wg2[j];
    gate[node] = s;
}

__global__ void pool_max_kernel(const int* __restrict__ batch, const float* __restrict__ gate,
                                float* __restrict__ gmax, int n) {
    int i = blockIdx.x * blockDim.x + threadIdx.x;
    if (i < n) atomicMaxF(&gmax[batch[i]], gate[i]);
}

__global__ void pool_sum_kernel(const int* __restrict__ batch, const float* __restrict__ gate,
                                const float* __restrict__ gmax, float* __restrict__ gsum, int n) {
    int i = blockIdx.x * blockDim.x + threadIdx.x;
    if (i < n) atomicAdd(&gsum[batch[i]], expf(gate[i] - gmax[batch[i]]));
}

__global__ __launch_bounds__(HID) void pool_scatter_kernel(
    const float* __restrict__ h, const int* __restrict__ batch, const float* __restrict__ gate,
    const float* __restrict__ gmax, const float* __restrict__ gsum, float* __restrict__ hg) {
    int n = blockIdx.x, f = threadIdx.x;
    int g = batch[n];
    float w = expf(gate[n] - gmax[g]) / fmaxf(gsum[g], 1e-16f);
    atomicAdd(&hg[(size_t)g * HID + f], h[(size_t)n * HID + f] * w);
}

// row LayerNorm + relu, in place. One block (HID threads) per row.
__global__ __launch_bounds__(HID) void ln_relu_kernel(float* __restrict__ X,
                                                      const float* __restrict__ g,
                                                      const float* __restrict__ b) {
    __shared__ float red[HID];
    int row = blockIdx.x, f = threadIdx.x;
    float v = X[(size_t)row * HID + f];
    red[f] = v;
    __syncthreads();
    for (int s = HID / 2; s > 0; s >>= 1) {
        if (f < s) red[f] += red[f + s];
        __syncthreads();
    }
    float mean = red[0] / (float)HID;
    __syncthreads();
    float dv = v - mean;
    red[f] = dv * dv;
    __syncthreads();
    for (int s = HID / 2; s > 0; s >>= 1) {
        if (f < s) red[f] += red[f + s];
        __syncthreads();
    }
    float var = red[0] / (float)HID;
    float y = dv * rsqrtf(var + EPSV) * g[f] + b[f];
    X[(size_t)row * HID + f] = fmaxf(y, 0.0f);
}

// ---------------------------------------------------------------------------
// Host-side orchestration
// ---------------------------------------------------------------------------
extern "C" void kernel_launch(void* const* d_in, const int* in_sizes, int n_in,
                              void* d_out, int out_size, void* d_ws, size_t ws_size,
                              hipStream_t stream) {
    (void)in_sizes; (void)n_in; (void)out_size; (void)ws_size;

    const float* x       = (const float*)d_in[0];
    const float* W_emb   = (const float*)d_in[1];
    const float* b_emb   = (const float*)d_in[2];
    const float* W_gcn   = (const float*)d_in[3];
    const float* W_gat   = (const float*)d_in[5];
    const float* a_src   = (const float*)d_in[6];
    const float* a_dst   = (const float*)d_in[7];
    const float* W_l     = (const float*)d_in[9];
    const float* W_r     = (const float*)d_in[11];
    const float* bn_g    = (const float*)d_in[12];
    const float* bn_b    = (const float*)d_in[13];
    const float* Wg1     = (const float*)d_in[14];
    const float* bg1     = (const float*)d_in[15];
    const float* Wg2     = (const float*)d_in[16];
    const float* bg2     = (const float*)d_in[17];
    const float* Wm1     = (const float*)d_in[18];
    const float* bm1     = (const float*)d_in[19];
    const float* ln_m_g  = (const float*)d_in[20];
    const float* ln_m_b  = (const float*)d_in[21];
    const float* Wm2     = (const float*)d_in[22];
    const float* bm2     = (const float*)d_in[23];
    const float* Wv1     = (const float*)d_in[24];
    const float* bv1     = (const float*)d_in[25];
    const float* ln_v_g  = (const float*)d_in[26];
    const float* ln_v_b  = (const float*)d_in[27];
    const float* Wv2     = (const float*)d_in[28];
    const float* bv2     = (const float*)d_in[29];
    const int*   ei      = (const int*)d_in[30];   // [2, E] flattened
    const int*   batch   = (const int*)d_in[31];   // [N]

    // Workspace layout (floats)
    float* ws = (float*)d_ws;
    const size_t NH = (size_t)N_NODES * HID;
    float* h      = ws;               ws += NH;          // node state
    float* bufA   = ws;               ws += NH;          // xt / nb_mean / gate hidden
    float* bufB   = ws;               ws += NH;          // conv output accumulator
    float* deg    = ws;               ws += N_NODES;     // also SAGE cnt
    float* dis    = ws;               ws += N_NODES;
    float* alsrc  = ws;               ws += (size_t)N_NODES * HEADS;
    float* aldst  = ws;               ws += (size_t)N_NODES * HEADS;
    float* segmax = ws;               ws += (size_t)N_NODES * HEADS;
    float* segsum = ws;               ws += (size_t)N_NODES * HEADS;
    float* gate   = ws;               ws += N_NODES;
    float* gmax   = ws;               ws += N_GR;
    float* gsum   = ws;               ws += N_GR;
    float* hg     = ws;               ws += (size_t)N_GR * HID;
    float* tbuf   = ws;               ws += (size_t)N_GR * HID;
    float* csum   = ws;               ws += HID;
    float* csumsq = ws;               ws += HID;
    float* cmu    = ws;               ws += HID;
    float* crs    = ws;               ws += HID;

    float* out_mu = (float*)d_out;                 // [512, 64]
    float* out_lv = out_mu + (size_t)N_GR * LAT;   // [512, 64]

    const int TOT = N_EDGES + N_NODES;  // entries with self loops
    const dim3 b128(128), b256(256);

    auto gemm = [&](const float* A, const float* B, float* C, const float* bias,
                    int M, int Nn, int K, int acc, int relu) {
        dim3 grid((M + 63) / 64, Nn / 64);
        gemm_wmma_f32<<<grid, b128, 0, stream>>>(A, B, C, bias, M, Nn, K, acc, relu);
    };
    auto fill = [&](float* p, size_t n, float v) {
        fill_kernel<<<1024, b256, 0, stream>>>(p, n, v);
    };
    auto bnstats = [&](const float* X) {
        fill(csum, HID, 0.0f);
        fill(csumsq, HID, 0.0f);
        colstat_partial_kernel<<<256, HID, 0, stream>>>(X, csum, csumsq, N_NODES);
        colstat_finalize_kernel<<<1, HID, 0, stream>>>(csum, csumsq, cmu, crs, N_NODES);
    };

    // ---- embedding: h = x @ W_emb + b_emb
    gemm(x, W_emb, h, b_emb, N_NODES, HID, N_FEAT, 0, 0);

    // ---- layer 0: GCN
    fill(deg, N_NODES, 0.0f);
    degree_kernel<<<(TOT + 255) / 256, b256, 0, stream>>>(ei, deg, N_EDGES, TOT);
    rsqrt_kernel<<<(N_NODES + 255) / 256, b256, 0, stream>>>(deg, dis, N_NODES);
    gemm(h, W_gcn, bufA, nullptr, N_NODES, HID, HID, 0, 0);
    fill(bufB, NH, 0.0f);
    gcn_scatter_kernel<<<TOT, HID, 0, stream>>>(ei, bufA, dis, bufB, N_EDGES);
    bnstats(bufB);
    bn_apply_residual_kernel<<<2048, b256, 0, stream>>>(h, bufB, cmu, crs, bn_g + 0 * HID, bn_b + 0 * HID, NH);

    // ---- layer 1: GAT (8 heads)
    gemm(h, W_gat, bufA, nullptr, N_NODES, HID, HID, 0, 0);
    gat_coeff_kernel<<<(N_NODES * HEADS + 255) / 256, b256, 0, stream>>>(bufA, a_src, a_dst, alsrc, aldst, N_NODES);
    fill(segmax, (size_t)N_NODES * HEADS, -INFINITY);
    gat_max_kernel<<<((size_t)TOT * HEADS + 255) / 256, b256, 0, stream>>>(ei, alsrc, aldst, segmax, N_EDGES, TOT);
    finite_fix_kernel<<<((size_t)N_NODES * HEADS + 255) / 256, b256, 0, stream>>>(segmax, (size_t)N_NODES * HEADS);
    fill(segsum, (size_t)N_NODES * HEADS, 0.0f);
    gat_sum_kernel<<<((size_t)TOT * HEADS + 255) / 256, b256, 0, stream>>>(ei, alsrc, aldst, segmax, segsum, N_EDGES, TOT);
    fill(bufB, NH, 0.0f);
    gat_scatter_kernel<<<TOT, HID, 0, stream>>>(ei, bufA, alsrc, aldst, segmax, segsum, bufB, N_EDGES);
    bnstats(bufB);
    bn_apply_residual_kernel<<<2048, b256, 0, stream>>>(h, bufB, cmu, crs, bn_g + 1 * HID, bn_b + 1 * HID, NH);

    // ---- layer 2: SAGE (mean aggregation, edges only)
    fill(deg, N_NODES, 0.0f);
    degree_kernel<<<(N_EDGES + 255) / 256, b256, 0, stream>>>(ei, deg, N_EDGES, N_EDGES);
    fill(bufA, NH, 0.0f);
    sage_scatter_kernel<<<N_EDGES, HID, 0, stream>>>(ei, h, bufA, N_EDGES);
    sage_mean_kernel<<<2048, b256, 0, stream>>>(bufA, deg, NH);
    gemm(bufA, W_l, bufB, nullptr, N_NODES, HID, HID, 0, 0);
    gemm(h, W_r, bufB, nullptr, N_NODES, HID, HID, 1, 0);   // accumulate
    bnstats(bufB);
    bn_apply_residual_kernel<<<2048, b256, 0, stream>>>(h, bufB, cmu, crs, bn_g + 2 * HID, bn_b + 2 * HID, NH);

    // ---- GlobalAttention pooling
    gemm(h, Wg1, bufA, bg1, N_NODES, HID / 2, HID, 0, 1);   // relu(h@Wg1+bg1) -> [N,128]
    gate_kernel<<<(N_NODES + 255) / 256, b256, 0, stream>>>(bufA, Wg2, bg2, gate, N_NODES);
    fill(gmax, N_GR, -INFINITY);
    pool_max_kernel<<<(N_NODES + 255) / 256, b256, 0, stream>>>(batch, gate, gmax, N_NODES);
    finite_fix_kernel<<<(N_GR + 255) / 256, b256, 0, stream>>>(gmax, N_GR);
    fill(gsum, N_GR, 0.0f);
    pool_sum_kernel<<<(N_NODES + 255) / 256, b256, 0, stream>>>(batch, gate, gmax, gsum, N_NODES);
    fill(hg, (size_t)N_GR * HID, 0.0f);
    pool_scatter_kernel<<<N_NODES, HID, 0, stream>>>(h, batch, gate, gmax, gsum, hg);

    // ---- heads: mu / logvar
    gemm(hg, Wm1, tbuf, bm1, N_GR, HID, HID, 0, 0);
    ln_relu_kernel<<<N_GR, HID, 0, stream>>>(tbuf, ln_m_g, ln_m_b);
    gemm(tbuf, Wm2, out_mu, bm2, N_GR, LAT, HID, 0, 0);

    gemm(hg, Wv1, tbuf, bv1, N_GR, HID, HID, 0, 0);
    ln_relu_kernel<<<N_GR, HID, 0, stream>>>(tbuf, ln_v_g, ln_v_b);
    gemm(tbuf, Wv2, out_lv, bv2, N_GR, LAT, HID, 0, 0);
}